// Channel_attention_60859686584686
// MI455X (gfx1250) — compile-verified
//
#include <hip/hip_runtime.h>
#include <hip/hip_bf16.h>

typedef __attribute__((ext_vector_type(2))) float v2f;
typedef __attribute__((ext_vector_type(4))) float v4f;
typedef __attribute__((ext_vector_type(8))) float v8f;

#define CCH 64
#define TT 2000
#define NF 8
#define CHUNK 64
#define NCHUNK 32      // 32*64 = 2048 >= 2000 (tail zero-padded)
#define PITCH 68       // LDS row pitch (floats): conflict-free strided reads
#define EPSV 1e-8f

__global__ __launch_bounds__(256)
void chan_attn_kernel(const float* __restrict__ x,
                      const float* __restrict__ w1, const float* __restrict__ b1,
                      const float* __restrict__ w2, const float* __restrict__ b2,
                      const float* __restrict__ gammap,
                      float* __restrict__ out)
{
    __shared__ float xbuf[CCH * PITCH];   // streamed X chunk [64 ch][64 t]
    __shared__ float attn[CCH * PITCH];   // energy -> attention [64][64]
    __shared__ float sred[4 * CCH];       // row-sum partials
    __shared__ float ssum[CCH];           // row sums s[c]

    const int tid  = threadIdx.x;
    const int lane = tid & 31;
    const int wave = tid >> 5;
    const int half = lane >> 4;           // f32 WMMA: lanes 16-31 hold K=2,3
    const int m16  = lane & 15;
    const int b    = blockIdx.x;

    const float* xb = x   + (size_t)b * CCH * TT;
    float*       ob = out + (size_t)b * CCH * TT;

    // ---- uniform filter contractions (F=8; becomes scalar loads) ----
    float s_ww = 0.f, s_wb2 = 0.f, s_b1w = 0.f, s_bb = 0.f;
    #pragma unroll
    for (int f = 0; f < NF; ++f) {
        float a1 = w1[f], a2 = w2[f], c1 = b1[f], c2 = b2[f];
        s_ww += a1 * a2;  s_wb2 += a1 * c2;
        s_b1w += c1 * a2; s_bb  += c1 * c2;
    }
    const float econst = (float)TT * s_bb;
    const float gamma  = gammap[0];

    // =========== Phase 1: Gram G = X X^T via v_wmma_f32_16x16x4_f32 ===========
    // 4x4 grid of 16x16 tiles; wave w owns tiles 2w and 2w+1
    v8f acc0 = {}; v8f acc1 = {};
    const int t0i = (2 * wave) >> 2,     t0j = (2 * wave) & 3;
    const int t1i = (2 * wave + 1) >> 2, t1j = (2 * wave + 1) & 3;

    float rs_part   = 0.f;               // row-sum partial
    const int rs_c  = tid & 63;
    const int rs_g  = tid >> 6;

    const int ld_c   = tid >> 2;         // cooperative chunk loader mapping
    const int ld_seg = tid & 3;

    for (int ch = 0; ch < NCHUNK; ++ch) {
        __syncthreads();
        // load X[:, ch*64 .. +63] -> LDS (b128 global loads, zero tail)
        {
            const int tg0 = ch * CHUNK + ld_seg * 16;
            #pragma unroll
            for (int q = 0; q < 4; ++q) {
                int t = tg0 + q * 4;
                v4f v;
                if (t + 3 < TT) {
                    v = *(const v4f*)(xb + ld_c * TT + t);
                } else {
                    v.x = (t + 0 < TT) ? xb[ld_c * TT + t + 0] : 0.f;
                    v.y = (t + 1 < TT) ? xb[ld_c * TT + t + 1] : 0.f;
                    v.z = (t + 2 < TT) ? xb[ld_c * TT + t + 2] : 0.f;
                    v.w = (t + 3 < TT) ? xb[ld_c * TT + t + 3] : 0.f;
                }
                *(v4f*)&xbuf[ld_c * PITCH + ld_seg * 16 + q * 4] = v;
            }
            if (ch + 1 < NCHUNK) {  // speculative L2 prefetch of next chunk
                int tn = (ch + 1) * CHUNK + ld_seg * 16;
                if (tn < TT) __builtin_prefetch(xb + ld_c * TT + tn, 0, 1);
            }
        }
        __syncthreads();

        // Gram accumulation: A = X[16i.., k..k+3], B = X[16j.., k..k+3]^T
        #pragma unroll
        for (int k = 0; k < CHUNK; k += 4) {
            const int kk = k + 2 * half;
            v2f a0 = *(const v2f*)&xbuf[(t0i * 16 + m16) * PITCH + kk];
            v2f b0 = *(const v2f*)&xbuf[(t0j * 16 + m16) * PITCH + kk];
            acc0 = __builtin_amdgcn_wmma_f32_16x16x4_f32(
                       false, a0, false, b0, (short)0, acc0, false, false);
            v2f a1 = *(const v2f*)&xbuf[(t1i * 16 + m16) * PITCH + kk];
            v2f b1v = *(const v2f*)&xbuf[(t1j * 16 + m16) * PITCH + kk];
            acc1 = __builtin_amdgcn_wmma_f32_16x16x4_f32(
                       false, a1, false, b1v, (short)0, acc1, false, false);
        }
        // row-sum partial over this chunk's t-slice
        #pragma unroll
        for (int q = 0; q < 16; ++q)
            rs_part += xbuf[rs_c * PITCH + rs_g * 16 + q];
    }
    sred[rs_g * CCH + rs_c] = rs_part;
    __syncthreads();
    if (tid < CCH)
        ssum[tid] = sred[tid] + sred[CCH + tid] + sred[2 * CCH + tid] + sred[3 * CCH + tid];
    __syncthreads();

    // =========== Phase 2: energy = affine(G); row minmax-norm + softmax ===========
    #pragma unroll
    for (int r = 0; r < 8; ++r) {
        int row = t0i * 16 + r + 8 * half;
        int col = t0j * 16 + m16;
        attn[row * PITCH + col] = s_ww * acc0[r] + s_wb2 * ssum[row] + s_b1w * ssum[col] + econst;
        row = t1i * 16 + r + 8 * half;
        col = t1j * 16 + m16;
        attn[row * PITCH + col] = s_ww * acc1[r] + s_wb2 * ssum[row] + s_b1w * ssum[col] + econst;
    }
    __syncthreads();

    // wave w normalizes rows 8w..8w+7 (wave32 shfl reductions over 64 cols)
    #pragma unroll
    for (int r = 0; r < 8; ++r) {
        const int row = wave * 8 + r;
        float v0 = attn[row * PITCH + lane];
        float v1 = attn[row * PITCH + lane + 32];
        float mn = fminf(v0, v1), mx = fmaxf(v0, v1);
        #pragma unroll
        for (int off = 16; off; off >>= 1) {
            mn = fminf(mn, __shfl_xor(mn, off, 32));
            mx = fmaxf(mx, __shfl_xor(mx, off, 32));
        }
        const float inv = 1.0f / (mx - mn + EPSV);
        // normalized values lie in [0,1] -> exp() is safe without max-shift
        float p0 = __expf((v0 - mn) * inv);
        float p1 = __expf((v1 - mn) * inv);
        float s  = p0 + p1;
        #pragma unroll
        for (int off = 16; off; off >>= 1) s += __shfl_xor(s, off, 32);
        const float rinv = 1.0f / s;
        attn[row * PITCH + lane]      = p0 * rinv;
        attn[row * PITCH + lane + 32] = p1 * rinv;
    }
    __syncthreads();

    // =========== Phase 3: out = gamma * (attn^T @ X) + X, streamed ===========
    for (int ch = 0; ch < NCHUNK; ++ch) {
        // reload chunk (hits L2; whole x fits in 192MB L2)
        {
            const int tg0 = ch * CHUNK + ld_seg * 16;
            #pragma unroll
            for (int q = 0; q < 4; ++q) {
                int t = tg0 + q * 4;
                v4f v;
                if (t + 3 < TT) {
                    v = *(const v4f*)(xb + ld_c * TT + t);
                } else {
                    v.x = (t + 0 < TT) ? xb[ld_c * TT + t + 0] : 0.f;
                    v.y = (t + 1 < TT) ? xb[ld_c * TT + t + 1] : 0.f;
                    v.z = (t + 2 < TT) ? xb[ld_c * TT + t + 2] : 0.f;
                    v.w = (t + 3 < TT) ? xb[ld_c * TT + t + 3] : 0.f;
                }
                *(v4f*)&xbuf[ld_c * PITCH + ld_seg * 16 + q * 4] = v;
            }
        }
        __syncthreads();

        // wave w computes output tiles 2w, 2w+1 of the 4(c) x 4(t) grid
        #pragma unroll
        for (int which = 0; which < 2; ++which) {
            const int tile = 2 * wave + which;
            const int ci = tile >> 2, tj = tile & 3;
            v8f o = {};
            #pragma unroll
            for (int k = 0; k < CCH; k += 4) {
                const int kk = k + 2 * half;
                v2f a;          // A[M=c, K=e] = attn[e][c]  (attn transposed)
                a.x = attn[kk * PITCH + ci * 16 + m16];
                a.y = attn[(kk + 1) * PITCH + ci * 16 + m16];
                v2f bb;         // B[K=e, N=t] = x[e][t]
                bb.x = xbuf[kk * PITCH + tj * 16 + m16];
                bb.y = xbuf[(kk + 1) * PITCH + tj * 16 + m16];
                o = __builtin_amdgcn_wmma_f32_16x16x4_f32(
                        false, a, false, bb, (short)0, o, false, false);
            }
            // residual + store: element (M = r + 8*half, N = m16)
            #pragma unroll
            for (int r = 0; r < 8; ++r) {
                const int c = ci * 16 + r + 8 * half;
                const int t = ch * CHUNK + tj * 16 + m16;
                if (t < TT)
                    ob[c * TT + t] = gamma * o[r] + xbuf[c * PITCH + tj * 16 + m16];
            }
        }
        __syncthreads();   // protect xbuf before next chunk overwrite
    }
}

extern "C" void kernel_launch(void* const* d_in, const int* in_sizes, int n_in,
                              void* d_out, int out_size, void* d_ws, size_t ws_size,
                              hipStream_t stream) {
    const float* x  = (const float*)d_in[0];
    const float* w1 = (const float*)d_in[1];
    const float* b1 = (const float*)d_in[2];
    const float* w2 = (const float*)d_in[3];
    const float* b2 = (const float*)d_in[4];
    const float* g  = (const float*)d_in[5];
    float* out = (float*)d_out;
    const int B = in_sizes[0] / (CCH * TT);   // 128
    chan_attn_kernel<<<B, 256, 0, stream>>>(x, w1, b1, w2, b2, g, out);
}